// HeteroGATModel_24739011625783
// MI455X (gfx1250) — compile-verified
//
#include <hip/hip_runtime.h>
#include <hip/hip_bf16.h>

typedef __attribute__((ext_vector_type(2))) float v2f;
typedef __attribute__((ext_vector_type(8))) float v8f;

#define HID 32
#define HEADS 4
#define HC 128            // HEADS*HID
#define NEG_SLOPE 0.2f

__device__ __forceinline__ float eluf(float x)   { return x > 0.f ? x : __expf(x) - 1.f; }
__device__ __forceinline__ float lreluf(float x) { return x > 0.f ? x : NEG_SLOPE * x; }

// float atomic max via integer sign trick (init value must be -inf)
__device__ __forceinline__ void atomicMaxFloat(float* addr, float val) {
  if (val >= 0.f) atomicMax((int*)addr, __float_as_int(val));
  else            atomicMin((unsigned int*)addr, __float_as_uint(val));
}

__global__ void fill_kernel(float* __restrict__ p, float v, long n) {
  long i = blockIdx.x * (long)blockDim.x + threadIdx.x;
  if (i < n) p[i] = v;
}

// WT[N,K] = transpose(W[K,N]) -- weights are KB-sized, done once up front
__global__ void transpose_kernel(const float* __restrict__ W, float* __restrict__ WT,
                                 int K, int N) {
  const int i = blockIdx.x * blockDim.x + threadIdx.x;
  if (i >= K * N) return;
  const int k = i / N, n = i % N;
  WT[(long)n * K + k] = W[i];
}

// out[M,N] = maybe_elu(A[M,K] @ W[K,N] + bias[N]) with V_WMMA_F32_16X16X4_F32.
// W is pre-transposed (WT[N,K]). K-chunks of WT are staged into LDS with
// global_load_async_to_lds_b128 (ASYNCcnt) and read back as ds_load_b64 B-fragments.
// One wave owns a 16-row tile; K loop is outermost so A streams exactly once.
// Requires M % 16 == 0 (50000 = 3125*16), CK % 4 == 0, K % CK == 0, N % 32 == 0, CK <= 64.
template <int K, int N, int CK, bool DO_ELU>
__global__ void wmma_gemm_kernel(const float* __restrict__ A,
                                 const float* __restrict__ WT,
                                 const float* __restrict__ bias,
                                 float* __restrict__ out, int M) {
  constexpr int NG     = N / 32;      // 32-column groups
  constexpr int LST    = 68;          // padded LDS row stride (floats): even + no 64-bank hotspot
  constexpr int NCHUNK = K / CK;
  constexpr int VECS   = N * CK / 4;  // float4 transfers per chunk
  __shared__ float lds[N * LST];

  const int  tid    = threadIdx.x;
  const int  wave   = tid >> 5;
  const int  tile   = blockIdx.x * (blockDim.x >> 5) + wave;
  const bool active = (tile * 16 < M);          // wave-uniform (EXEC stays all-1s)
  const int  lane   = tid & 31;
  const int  hl     = lane >> 4;                // wave half: K pair 0/1 vs 2/3
  const int  lm     = lane & 15;
  const long arow   = (long)(tile * 16 + lm) * K;

  v8f acc[NG][2];
  #pragma unroll
  for (int g = 0; g < NG; ++g) { acc[g][0] = {}; acc[g][1] = {}; }

  for (int c = 0; c < NCHUNK; ++c) {
    const int k0 = c * CK;
    __syncthreads();                            // previous chunk fully consumed
    for (int v = tid; v < VECS; v += 256) {     // cooperative async stage of WT chunk
      const int n  = v / (CK / 4);
      const int kq = v % (CK / 4);
      const unsigned loff = (unsigned)(size_t)&lds[n * LST + kq * 4];  // low 32 bits = LDS offset
      const void* gsrc = (const void*)(WT + (long)n * K + k0 + kq * 4);
      asm volatile("global_load_async_to_lds_b128 %0, %1, off"
                   :: "v"(loff), "v"(gsrc) : "memory");
    }
    asm volatile("s_wait_asynccnt 0x0" ::: "memory");
    __syncthreads();                            // chunk visible to all waves
    if (active) {
      if (c + 1 < NCHUNK) __builtin_prefetch(A + arow + k0 + CK, 0, 1);
      for (int k = 0; k < CK; k += 4) {
        const int kk = k + 2 * hl;
        // A fragment: lanes 0-15 hold (K=k,k+1), lanes 16-31 hold (K=k+2,k+3), M = lm
        v2f a = *(const v2f*)(A + arow + k0 + kk);
        #pragma unroll
        for (int g = 0; g < NG; ++g) {
          const int n0 = g * 32 + lm;
          v2f b0 = *(const v2f*)&lds[n0 * LST + kk];          // ds_load_b64
          v2f b1 = *(const v2f*)&lds[(n0 + 16) * LST + kk];
          acc[g][0] = __builtin_amdgcn_wmma_f32_16x16x4_f32(false, a, false, b0, (short)0, acc[g][0], false, false);
          acc[g][1] = __builtin_amdgcn_wmma_f32_16x16x4_f32(false, a, false, b1, (short)0, acc[g][1], false, false);
        }
      }
    }
  }
  if (!active) return;
  // C/D layout: VGPR i, lanes 0-15 -> M=i, lanes 16-31 -> M=8+i, N = lane%16
  #pragma unroll
  for (int g = 0; g < NG; ++g) {
    const int nc0 = g * 32 + lm;
    #pragma unroll
    for (int i = 0; i < 8; ++i) {
      const int m = tile * 16 + i + 8 * hl;
      float r0 = acc[g][0][i] + bias[nc0];
      float r1 = acc[g][1][i] + bias[nc0 + 16];
      if (DO_ELU) { r0 = eluf(r0); r1 = eluf(r1); }
      out[(long)m * N + nc0]      = r0;
      out[(long)m * N + nc0 + 16] = r1;
    }
  }
}

// One wave per edge. lane l: head h=l>>3, channels (l&7)*4..+3.
// s[e,h] = att[h,:] . leakyrelu(xl[src,h,:] + xr[dst,h,:]); segment-max into emax[dst,h].
__global__ void edge_score_kernel(const int* __restrict__ ei, int E,
                                  const float* __restrict__ xl,
                                  const float* __restrict__ xr,
                                  const float* __restrict__ att,
                                  float* __restrict__ s,
                                  float* __restrict__ emax) {
  const int e = (int)((blockIdx.x * (long)blockDim.x + threadIdx.x) >> 5);
  if (e >= E) return;
  const int lane = threadIdx.x & 31;
  const int src = ei[e];
  const int dst = ei[E + e];
  const float4 a = *(const float4*)(xl + (long)src * HC + lane * 4);
  const float4 c = *(const float4*)(xr + (long)dst * HC + lane * 4);
  const float4 w = *(const float4*)(att + lane * 4);   // att[4,32] flat == lane*4
  float p = lreluf(a.x + c.x) * w.x + lreluf(a.y + c.y) * w.y +
            lreluf(a.z + c.z) * w.z + lreluf(a.w + c.w) * w.w;
  p += __shfl_xor(p, 1, 8);
  p += __shfl_xor(p, 2, 8);
  p += __shfl_xor(p, 4, 8);
  if ((lane & 7) == 0) {
    const int h = lane >> 3;
    s[(long)e * HEADS + h] = p;
    atomicMaxFloat(emax + (long)dst * HEADS + h, p);
  }
}

// ee = exp(s - emax[dst]); s <- ee; denom[dst] += ee
__global__ void edge_softmax_kernel(const int* __restrict__ ei, int E,
                                    float* __restrict__ s,
                                    const float* __restrict__ emax,
                                    float* __restrict__ denom) {
  const long t = blockIdx.x * (long)blockDim.x + threadIdx.x;
  if (t >= (long)E * HEADS) return;
  const int e = (int)(t >> 2), h = (int)(t & 3);
  const int dst = ei[E + e];
  const float ee = __expf(s[t] - emax[(long)dst * HEADS + h]);
  s[t] = ee;
  atomicAdd(denom + (long)dst * HEADS + h, ee);
}

// accum[dst,h,:] += (ee/denom) * xl[src,h,:]   (one wave per edge, float4 per lane)
__global__ void edge_aggregate_kernel(const int* __restrict__ ei, int E,
                                      const float* __restrict__ xl,
                                      const float* __restrict__ s,
                                      const float* __restrict__ denom,
                                      float* __restrict__ accum) {
  const int e = (int)((blockIdx.x * (long)blockDim.x + threadIdx.x) >> 5);
  if (e >= E) return;
  const int lane = threadIdx.x & 31;
  const int h = lane >> 3;
  const int src = ei[e];
  const int dst = ei[E + e];
  const float alpha = s[(long)e * HEADS + h] / (denom[(long)dst * HEADS + h] + 1e-16f);
  const float4 a = *(const float4*)(xl + (long)src * HC + lane * 4);
  float* o = accum + (long)dst * HC + lane * 4;
  atomicAdd(o + 0, alpha * a.x);
  atomicAdd(o + 1, alpha * a.y);
  atomicAdd(o + 2, alpha * a.z);
  atomicAdd(o + 3, alpha * a.w);
}

__global__ void bias_elu_kernel(const float* __restrict__ in,
                                const float* __restrict__ bias,
                                float* __restrict__ out, long total) {
  const long i = blockIdx.x * (long)blockDim.x + threadIdx.x;
  if (i >= total) return;
  out[i] = eluf(in[i] + bias[i & (HC - 1)]);
}

// out[n,0:2] = h[n,:] @ W_out[128,2] + b_out
__global__ void out_head_kernel(const float* __restrict__ h,
                                const float* __restrict__ W,
                                const float* __restrict__ b,
                                float* __restrict__ out, int M) {
  const int n = blockIdx.x * blockDim.x + threadIdx.x;
  if (n >= M) return;
  float a0 = b[0], a1 = b[1];
  const float4* hp = (const float4*)(h + (long)n * HC);
  #pragma unroll 8
  for (int q = 0; q < HC / 4; ++q) {
    const float4 v = hp[q];
    const float* w = W + q * 8;
    a0 += v.x * w[0] + v.y * w[2] + v.z * w[4] + v.w * w[6];
    a1 += v.x * w[1] + v.y * w[3] + v.z * w[5] + v.w * w[7];
  }
  out[n * 2]     = a0;
  out[n * 2 + 1] = a1;
}

extern "C" void kernel_launch(void* const* d_in, const int* in_sizes, int n_in,
                              void* d_out, int out_size, void* d_ws, size_t ws_size,
                              hipStream_t stream) {
  const float* x_news   = (const float*)d_in[0];
  const float* x_user   = (const float*)d_in[1];
  const int*   ei_un    = (const int*)  d_in[2];   // (user -> news): row0 = src(user), row1 = dst(news)
  const float* W_news   = (const float*)d_in[4];
  const float* b_news   = (const float*)d_in[5];
  const float* W_user   = (const float*)d_in[6];
  const float* b_user   = (const float*)d_in[7];
  const float* Wl1      = (const float*)d_in[8];
  const float* bl1      = (const float*)d_in[9];
  const float* Wr1      = (const float*)d_in[10];
  const float* br1      = (const float*)d_in[11];
  const float* att1     = (const float*)d_in[12];
  const float* bias1    = (const float*)d_in[13];
  // conv2 (indices 3, 14..19) is dead code: its result h_user never reaches the output.
  const float* W_out    = (const float*)d_in[20];
  const float* b_out    = (const float*)d_in[21];
  float* outp = (float*)d_out;

  const int Mn = in_sizes[0] / 768;   // 50000 news
  const int Mu = in_sizes[1] / 300;   // 50000 users
  const int E  = in_sizes[2] / 2;     // 600000 edges

  // workspace carve-out (256B-aligned slices)
  char* p = (char*)d_ws;
  auto take = [&](long nfloats) {
    float* r = (float*)p;
    p += ((nfloats * sizeof(float)) + 255) & ~(size_t)255;
    return r;
  };
  float* wt_news = take(768L * HID);       // W_news^T
  float* wt_user = take(300L * HID);       // W_user^T
  float* wtl1    = take((long)HID * HC);   // Wl1^T
  float* wtr1    = take((long)HID * HC);   // Wr1^T
  float* h_news  = take((long)Mn * HID);   // elu(x_news @ W_news + b)
  float* h_user  = take((long)Mu * HID);   // elu(x_user @ W_user + b)
  float* xl1     = take((long)Mu * HC);    // h_user @ Wl1 + bl1
  float* xr1     = take((long)Mn * HC);    // h_news @ Wr1 + br1  (later reused as accum)
  float* sbuf    = take((long)E * HEADS);  // edge scores -> exp weights
  float* emax    = take((long)Mn * HEADS);
  float* denom   = take((long)Mn * HEADS);
  float* h_news2 = take((long)Mn * HC);    // elu(conv1 output + bias1)
  float* accum   = xr1;                    // alias: xr1 dead after edge_score

  const int TB = 256;
  auto gemm_grid = [](int M) { return ((M + 15) / 16 + 7) / 8; };

  // transpose the (tiny) weight matrices once
  transpose_kernel<<<(768 * HID + TB - 1) / TB, TB, 0, stream>>>(W_news, wt_news, 768, HID);
  transpose_kernel<<<(300 * HID + TB - 1) / TB, TB, 0, stream>>>(W_user, wt_user, 300, HID);
  transpose_kernel<<<(HID * HC + TB - 1) / TB, TB, 0, stream>>>(Wl1, wtl1, HID, HC);
  transpose_kernel<<<(HID * HC + TB - 1) / TB, TB, 0, stream>>>(Wr1, wtr1, HID, HC);

  // node-type input projections (f32 WMMA, async-LDS staged weights) + ELU
  wmma_gemm_kernel<768, HID, 64, true><<<gemm_grid(Mn), TB, 0, stream>>>(x_news, wt_news, b_news, h_news, Mn);
  wmma_gemm_kernel<300, HID, 60, true><<<gemm_grid(Mu), TB, 0, stream>>>(x_user, wt_user, b_user, h_user, Mu);

  // GATv2 conv1 linear parts: xl = src proj (users), xr = dst proj (news)
  wmma_gemm_kernel<HID, HC, 32, false><<<gemm_grid(Mu), TB, 0, stream>>>(h_user, wtl1, bl1, xl1, Mu);
  wmma_gemm_kernel<HID, HC, 32, false><<<gemm_grid(Mn), TB, 0, stream>>>(h_news, wtr1, br1, xr1, Mn);

  // segment softmax state
  const long segN = (long)Mn * HEADS;
  fill_kernel<<<(int)((segN + TB - 1) / TB), TB, 0, stream>>>(emax, -__builtin_inff(), segN);
  fill_kernel<<<(int)((segN + TB - 1) / TB), TB, 0, stream>>>(denom, 0.f, segN);

  const int edgeWaveBlocks = (int)(((long)E * 32 + TB - 1) / TB);
  edge_score_kernel<<<edgeWaveBlocks, TB, 0, stream>>>(ei_un, E, xl1, xr1, att1, sbuf, emax);

  // xr1 is now dead -> zero it as the aggregation accumulator
  const long accN = (long)Mn * HC;
  fill_kernel<<<(int)((accN + TB - 1) / TB), TB, 0, stream>>>(accum, 0.f, accN);

  edge_softmax_kernel<<<(int)(((long)E * HEADS + TB - 1) / TB), TB, 0, stream>>>(ei_un, E, sbuf, emax, denom);
  edge_aggregate_kernel<<<edgeWaveBlocks, TB, 0, stream>>>(ei_un, E, xl1, sbuf, denom, accum);

  // h_news2 = elu(accum + bias1)
  bias_elu_kernel<<<(int)((accN + TB - 1) / TB), TB, 0, stream>>>(accum, bias1, h_news2, accN);

  // final head on news nodes
  out_head_kernel<<<(Mn + TB - 1) / TB, TB, 0, stream>>>(h_news2, W_out, b_out, outp, Mn);
}